// VolumeDCTPatchLayer_37452114821754
// MI455X (gfx1250) — compile-verified
//
#include <hip/hip_runtime.h>
#include <hip/hip_bf16.h>

// CDNA5 / gfx1250: wave32, WMMA f32 16x16x4 native path.
typedef __attribute__((ext_vector_type(2))) float v2f;
typedef __attribute__((ext_vector_type(8))) float v8f;

#define NPATCH 8192            // 4*8*16*16 (b,c,nh,nw)
#define ROW_I 9216             // stride of i index (96*96) in elements
#define ROW_J 96               // stride of j index in elements
#define BASE_NH 55296          // 6*96*96
#define BASE_NW 576            // 6*96
#define BASE_BC 884736         // 96^3

// Orthonormal 6-point DCT-II matrix, rows padded to 8 with zeros.
// Rows 6,7 are zero so that G rows m>=36 (a=6,7) vanish automatically.
__device__ __constant__ float DCT6[8][6] = {
    { 0.408248290463863f,  0.408248290463863f,  0.408248290463863f,
      0.408248290463863f,  0.408248290463863f,  0.408248290463863f },
    { 0.557677535825205f,  0.408248290463863f,  0.149429245361342f,
     -0.149429245361342f, -0.408248290463863f, -0.557677535825205f },
    { 0.5f,                0.0f,               -0.5f,
     -0.5f,                0.0f,                0.5f },
    { 0.408248290463863f, -0.408248290463863f, -0.408248290463863f,
      0.408248290463863f,  0.408248290463863f, -0.408248290463863f },
    { 0.288675134594813f, -0.577350269189626f,  0.288675134594813f,
      0.288675134594813f, -0.577350269189626f,  0.288675134594813f },
    { 0.149429245361342f, -0.408248290463863f,  0.557677535825205f,
     -0.557677535825205f,  0.408248290463863f, -0.149429245361342f },
    { 0.f, 0.f, 0.f, 0.f, 0.f, 0.f },
    { 0.f, 0.f, 0.f, 0.f, 0.f, 0.f }
};

__global__ __launch_bounds__(256)
void VolumeDCTPatchLayer_kernel(const float* __restrict__ x,
                                float* __restrict__ out) {
    const int lane  = threadIdx.x & 31;
    const int wave  = threadIdx.x >> 5;
    const int patch = blockIdx.x * 8 + wave;   // wave-uniform
    if (patch >= NPATCH) return;               // uniform exit: EXEC stays all-1s

    // patch -> (b*8+c, nh, nw)
    const int nw = patch & 15;
    const int nh = (patch >> 4) & 15;
    const int bc = patch >> 8;
    const int base = bc * BASE_BC + nh * BASE_NH + nw * BASE_NW;

    const int  n  = lane & 15;            // column (d) within 16-wide tile
    const bool hi = (lane & 16) != 0;     // upper lane half (K offset +2 / row +8)

    // ---- A operand: constant Kronecker-DCT matrix G = M (x) M -------------
    // A tile (rt,kc): lane l, vgpr v -> G[16*rt + (l&15)][4*kc + 2*(l>>4) + v]
    // Per (rt, lane): only two table rows needed; all column indices are
    // compile-time, the lane-half pick is a single v_cndmask.
    v2f A[3][9];
#pragma unroll
    for (int rt = 0; rt < 3; ++rt) {
        const int m = rt * 16 + (lane & 15);
        const int a = m / 6;               // 0..7 (rows 6,7 are zero pads)
        const int u = m - a * 6;           // 0..5
        float ra[6], ru[6];
#pragma unroll
        for (int t = 0; t < 6; ++t) { ra[t] = DCT6[a][t]; ru[t] = DCT6[u][t]; }
#pragma unroll
        for (int kc = 0; kc < 9; ++kc) {
            const int k0 = kc * 4;         // kk=0 half: k0,k0+1 ; kk=2: k0+2,k0+3
            const float v0 = ra[(k0    ) / 6] * ru[(k0    ) % 6];
            const float v1 = ra[(k0 + 1) / 6] * ru[(k0 + 1) % 6];
            const float v2 = ra[(k0 + 2) / 6] * ru[(k0 + 2) % 6];
            const float v3 = ra[(k0 + 3) / 6] * ru[(k0 + 3) % 6];
            v2f aop;
            aop.x = hi ? v2 : v0;
            aop.y = hi ? v3 : v1;
            A[rt][kc] = aop;
        }
    }

    // ---- B-operand load offsets: select between two compile-time values ----
    int offX[9], offY[9];
#pragma unroll
    for (int kc = 0; kc < 9; ++kc) {
        const int k0 = kc * 4;
        const int o0 = ((k0    ) / 6) * ROW_I + ((k0    ) % 6) * ROW_J;
        const int o1 = ((k0 + 1) / 6) * ROW_I + ((k0 + 1) % 6) * ROW_J;
        const int o2 = ((k0 + 2) / 6) * ROW_I + ((k0 + 2) % 6) * ROW_J;
        const int o3 = ((k0 + 3) / 6) * ROW_I + ((k0 + 3) % 6) * ROW_J;
        offX[kc] = hi ? o2 : o0;
        offY[kc] = hi ? o3 : o1;
    }

    // ---- Warm WGP$ for the whole 36-row x 96-float patch (global_prefetch_b8)
#pragma unroll
    for (int pr = 0; pr < 4; ++pr) {
        const int t = pr * 32 + lane;
        if (t < 108) {
            const int k2 = t / 3, seg = t - k2 * 3;
            const int i2 = k2 / 6, j2 = k2 - i2 * 6;
            __builtin_prefetch(x + base + i2 * ROW_I + j2 * ROW_J + seg * 32, 0, 3);
        }
    }

    // ---- Stream 6 d-tiles of 16; each input element read exactly once ----
#pragma unroll 1
    for (int nt = 0; nt < 6; ++nt) {
        const int d = nt * 16 + n;

        // B operand: lane l, vgpr v -> X[k = 4*kc + 2*(l>>4) + v][d]
        v2f Bv[9];
#pragma unroll
        for (int kc = 0; kc < 9; ++kc) {
            v2f b;
            b.x = x[base + offX[kc] + d];
            b.y = x[base + offY[kc] + d];
            Bv[kc] = b;
        }

#pragma unroll
        for (int rt = 0; rt < 3; ++rt) {
            v8f acc = {0.f, 0.f, 0.f, 0.f, 0.f, 0.f, 0.f, 0.f};
#pragma unroll
            for (int kc = 0; kc < 9; ++kc) {
                // D = A x B + C  (v_wmma_f32_16x16x4_f32)
                acc = __builtin_amdgcn_wmma_f32_16x16x4_f32(
                    /*neg_a=*/false, A[rt][kc],
                    /*neg_b=*/false, Bv[kc],
                    /*c_mod=*/(short)0, acc,
                    /*reuse_a=*/false, /*reuse_b=*/false);
            }
            // D layout: vgpr r, lane l -> row 16*rt + 8*(l>>4) + r, col l&15.
            // Store offset / validity chosen between two compile-time variants.
#pragma unroll
            for (int r = 0; r < 8; ++r) {
                const int  mlo = rt * 16 + r;
                const int  mhi = mlo + 8;
                const bool vlo = (mlo < 36), vhi = (mhi < 36);
                if (!vlo && !vhi) continue;          // compile-time dead rows
                const int olo = (mlo / 6) * ROW_I + (mlo % 6) * ROW_J;
                const int ohi = (mhi / 6) * ROW_I + (mhi % 6) * ROW_J;
                const int off = hi ? ohi : olo;
                const bool valid = hi ? vhi : vlo;
                if (valid) out[base + off + d] = acc[r];
            }
        }
    }
}

extern "C" void kernel_launch(void* const* d_in, const int* in_sizes, int n_in,
                              void* d_out, int out_size, void* d_ws, size_t ws_size,
                              hipStream_t stream) {
    const float* x = (const float*)d_in[0];
    float* out = (float*)d_out;
    dim3 grid(NPATCH / 8);   // 1024 workgroups, one patch per wave
    dim3 block(256);         // 8 wave32 per workgroup
    VolumeDCTPatchLayer_kernel<<<grid, block, 0, stream>>>(x, out);
}